// SelfAttentionBlock_33767032881971
// MI455X (gfx1250) — compile-verified
//
#include <hip/hip_runtime.h>
#include <hip/hip_bf16.h>
#include <math.h>

typedef float v2f __attribute__((ext_vector_type(2)));
typedef float v8f __attribute__((ext_vector_type(8)));

#define C128 128
#define ENC_NEG_INF 0x007FFFFFu

// ---------- order-preserving float<->uint encoding for atomicMax ----------
__device__ __forceinline__ unsigned fenc(float f) {
    unsigned u = __float_as_uint(f);
    return (u & 0x80000000u) ? ~u : (u | 0x80000000u);
}
__device__ __forceinline__ float fdec(unsigned u) {
    return (u & 0x80000000u) ? __uint_as_float(u & 0x7FFFFFFFu)
                             : __uint_as_float(~u);
}

// ---------- init: out_agg = bias_gat broadcast; denom=0; maxenc=-inf ----------
__global__ void init_node_kernel(float* __restrict__ buf1,
                                 const float* __restrict__ bias_gat,
                                 float* __restrict__ denom,
                                 unsigned* __restrict__ maxenc,
                                 int N, int NC) {
    int i = blockIdx.x * blockDim.x + threadIdx.x;
    if (i < NC) buf1[i] = bias_gat[i & (C128 - 1)];
    if (i < N) { denom[i] = 0.0f; maxenc[i] = ENC_NEG_INF; }
}

__global__ void zero_stats_kernel(float* __restrict__ stats, int n) {
    int i = blockIdx.x * blockDim.x + threadIdx.x;
    if (i < n) stats[i] = 0.0f;
}

// ---------- WMMA fp32 GEMM: Cmat[nrows x 128] = A[nrows x 128] * B[128 x 128] (+bias) ----------
__global__ __launch_bounds__(256)
void gemm_nc128_kernel(const float* __restrict__ A, const float* __restrict__ B,
                       const float* __restrict__ bias, float* __restrict__ Cmat,
                       int nrows) {
    __shared__ float lA[16 * C128];           // 8 KB A tile
    const int tid  = threadIdx.x;
    const int wave = tid >> 5;                // 0..7  (wave32)
    const int lane = tid & 31;
    const int row0 = blockIdx.x * 16;
    if (row0 >= nrows) return;

    // cooperative A-tile load: 2048 floats = 512 float4 across 256 threads
    {
        const float4* src = (const float4*)(A + (size_t)row0 * C128);
        float4* dst = (float4*)lA;
        dst[tid]       = src[tid];
        dst[tid + 256] = src[tid + 256];
    }
    __syncthreads();

    const int n0   = wave * 16;               // this wave's 16-column tile
    const int rc   = lane & 15;               // A row / B column within tile
    const int kk   = (lane >> 4) * 2;         // K sub-offset per lane-half
    const int half = lane >> 4;

    v8f acc = {};
#pragma unroll
    for (int k = 0; k < C128; k += 4) {
        v2f a, b;
        // A fragment 16x4 f32: lanes0-15 hold K=k,k+1; lanes16-31 hold K=k+2,k+3
        a.x = lA[rc * C128 + k + kk];
        a.y = lA[rc * C128 + k + kk + 1];
        // B fragment 4x16 f32: row K striped across lanes (N = lane&15)
        b.x = B[(size_t)(k + kk) * C128 + n0 + rc];
        b.y = B[(size_t)(k + kk + 1) * C128 + n0 + rc];
        acc = __builtin_amdgcn_wmma_f32_16x16x4_f32(
                  false, a, false, b, (short)0, acc, false, false);
    }

    // C/D layout: VGPR r -> M=r (lanes0-15) / M=r+8 (lanes16-31), N=lane&15
#pragma unroll
    for (int r = 0; r < 8; ++r) {
        int m   = r + half * 8;
        int col = n0 + rc;
        float v = acc[r];
        if (bias) v += bias[col];
        Cmat[(size_t)(row0 + m) * C128 + col] = v;
    }
}

// ---------- per-node attention scores: one wave (32 lanes) per node row ----------
__global__ __launch_bounds__(256)
void attn_scores_kernel(const float* __restrict__ xl,
                        const float* __restrict__ att_src,
                        const float* __restrict__ att_dst,
                        float* __restrict__ a_src, float* __restrict__ a_dst,
                        int N) {
    int gw   = (blockIdx.x * blockDim.x + threadIdx.x) >> 5;
    int lane = threadIdx.x & 31;
    if (gw >= N) return;
    const float4 v  = ((const float4*)(xl + (size_t)gw * C128))[lane];
    const float4 as = ((const float4*)att_src)[lane];
    const float4 ad = ((const float4*)att_dst)[lane];
    float ss = v.x * as.x + v.y * as.y + v.z * as.z + v.w * as.w;
    float sd = v.x * ad.x + v.y * ad.y + v.z * ad.z + v.w * ad.w;
#pragma unroll
    for (int off = 16; off > 0; off >>= 1) {
        ss += __shfl_xor(ss, off, 32);
        sd += __shfl_xor(sd, off, 32);
    }
    if (lane == 0) { a_src[gw] = ss; a_dst[gw] = sd; }
}

// ---------- edge pass A: leaky_relu + segment max ----------
__global__ void edge_max_kernel(const int* __restrict__ srcI, const int* __restrict__ dstI,
                                const float* __restrict__ a_src, const float* __restrict__ a_dst,
                                float* __restrict__ e_val, unsigned* __restrict__ maxenc,
                                int E, int etot) {
    int i = blockIdx.x * blockDim.x + threadIdx.x;
    if (i >= etot) return;
    int s, d;
    if (i < E) { s = srcI[i]; d = dstI[i]; } else { s = d = i - E; }
    float e = a_src[s] + a_dst[d];
    e = e > 0.0f ? e : 0.2f * e;            // leaky_relu, slope 0.2
    e_val[i] = e;
    atomicMax(&maxenc[d], fenc(e));
}

// ---------- edge pass B: exp(e - max) + segment sum ----------
__global__ void edge_exp_kernel(const int* __restrict__ dstI,
                                const unsigned* __restrict__ maxenc,
                                float* __restrict__ e_val, float* __restrict__ denom,
                                int E, int etot) {
    int i = blockIdx.x * blockDim.x + threadIdx.x;
    if (i >= etot) return;
    int d = (i < E) ? dstI[i] : (i - E);
    float w = __expf(e_val[i] - fdec(maxenc[d]));
    e_val[i] = w;
    atomicAdd(&denom[d], w);
}

// ---------- scatter-aggregate: 32 lanes per edge, float4 per lane ----------
__global__ __launch_bounds__(256)
void scatter_kernel(const int* __restrict__ srcI, const int* __restrict__ dstI,
                    const float* __restrict__ e_val, const float* __restrict__ denom,
                    const float* __restrict__ xl, float* __restrict__ out,
                    int E, int etot) {
    long long idx = (long long)blockIdx.x * blockDim.x + threadIdx.x;
    int edge = (int)(idx >> 5);
    int lane = (int)(idx & 31);
    if (edge >= etot) return;
    int s, d;
    if (edge < E) { s = srcI[edge]; d = dstI[edge]; } else { s = d = edge - E; }
    float coef = e_val[edge] / (denom[d] + 1e-16f);
    const float* arow = xl + (size_t)s * C128;
    __builtin_prefetch(arow + lane * 4, 0, 0);      // global_prefetch_b8
    const float4 v = ((const float4*)arow)[lane];
    float* o = out + (size_t)d * C128 + lane * 4;
    atomicAdd(o + 0, coef * v.x);
    atomicAdd(o + 1, coef * v.y);
    atomicAdd(o + 2, coef * v.z);
    atomicAdd(o + 3, coef * v.w);
}

// ---------- BN stats: two-level reduction (LDS partials -> global atomics) ----------
__global__ __launch_bounds__(256)
void bn_stats_kernel(const float* __restrict__ h, float* __restrict__ sum,
                     float* __restrict__ sumsq, int N) {
    __shared__ float s_sum[C128];
    __shared__ float s_sq[C128];
    int tid = threadIdx.x;
    if (tid < C128) { s_sum[tid] = 0.0f; s_sq[tid] = 0.0f; }
    __syncthreads();
    int c    = tid & (C128 - 1);
    int half = tid >> 7;
    int rpb  = (N + (int)gridDim.x - 1) / (int)gridDim.x;
    int r0   = (int)blockIdx.x * rpb;
    int r1   = r0 + rpb; if (r1 > N) r1 = N;
    float ls = 0.0f, lq = 0.0f;
    for (int r = r0 + half; r < r1; r += 2) {
        float v = h[(size_t)r * C128 + c];
        ls += v; lq += v * v;
    }
    atomicAdd(&s_sum[c], ls);
    atomicAdd(&s_sq[c], lq);
    __syncthreads();
    if (tid < C128) {
        atomicAdd(&sum[tid],   s_sum[tid]);
        atomicAdd(&sumsq[tid], s_sq[tid]);
    }
}

// ---------- BN finalize: per-channel scale/shift ----------
__global__ void bn_finalize_kernel(const float* __restrict__ sum, const float* __restrict__ sumsq,
                                   const float* __restrict__ g, const float* __restrict__ be,
                                   float* __restrict__ scale, float* __restrict__ shift, int N) {
    int c = threadIdx.x;
    if (c >= C128) return;
    float inv  = 1.0f / (float)N;
    float mu   = sum[c] * inv;
    float var  = sumsq[c] * inv - mu * mu;
    float rstd = rsqrtf(var + 1e-5f);
    float sc   = g[c] * rstd;
    scale[c] = sc;
    shift[c] = be[c] - mu * sc;
}

// ---------- BN apply + ReLU (+ optional residual) ----------
__global__ void bn_apply_kernel(const float* __restrict__ h, const float* __restrict__ scale,
                                const float* __restrict__ shift, const float* __restrict__ residual,
                                float* __restrict__ out, int n) {
    int i = blockIdx.x * blockDim.x + threadIdx.x;
    if (i >= n) return;
    int c = i & (C128 - 1);
    float v = h[i] * scale[c] + shift[c];
    v = v > 0.0f ? v : 0.0f;
    if (residual) v += residual[i];
    out[i] = v;
}

extern "C" void kernel_launch(void* const* d_in, const int* in_sizes, int n_in,
                              void* d_out, int out_size, void* d_ws, size_t ws_size,
                              hipStream_t stream) {
    const float* x        = (const float*)d_in[0];
    const int*   ei       = (const int*)d_in[1];      // int32 (JAX x64 disabled)
    const float* W        = (const float*)d_in[2];
    const float* att_src  = (const float*)d_in[3];
    const float* att_dst  = (const float*)d_in[4];
    const float* bias_gat = (const float*)d_in[5];
    const float* W1 = (const float*)d_in[6];
    const float* b1 = (const float*)d_in[7];
    const float* g1 = (const float*)d_in[8];
    const float* be1 = (const float*)d_in[9];
    const float* W2 = (const float*)d_in[10];
    const float* b2 = (const float*)d_in[11];
    const float* g2 = (const float*)d_in[12];
    const float* be2 = (const float*)d_in[13];
    float* out = (float*)d_out;

    const int N    = in_sizes[0] / C128;       // 50000
    const int E    = in_sizes[1] / 2;          // 800000
    const int etot = E + N;
    const int NC   = N * C128;
    const int* srcI = ei;
    const int* dstI = ei + E;

    // ---- workspace carve-up (floats) ----
    float* base = (float*)d_ws;
    size_t off = 0;
    float*    xl     = base + off; off += (size_t)NC;
    float*    buf1   = base + off; off += (size_t)NC;   // GAT out / normalized h
    float*    buf2   = base + off; off += (size_t)NC;   // pre-BN activations
    float*    a_src  = base + off; off += (size_t)N;
    float*    a_dst  = base + off; off += (size_t)N;
    float*    denom  = base + off; off += (size_t)N;
    unsigned* maxenc = (unsigned*)(base + off); off += (size_t)N;
    float*    e_val  = base + off; off += (size_t)etot;
    float*    stats  = base + off; off += 8 * C128;
    float *sum1 = stats, *sq1 = stats + C128, *scale1 = stats + 2*C128, *shift1 = stats + 3*C128;
    float *sum2 = stats + 4*C128, *sq2 = stats + 5*C128, *scale2 = stats + 6*C128, *shift2 = stats + 7*C128;
    (void)ws_size; (void)n_in; (void)out_size;

    const int T = 256;
    // 0) re-init everything every call (graph replays must be deterministic)
    init_node_kernel<<<(NC + T - 1) / T, T, 0, stream>>>(buf1, bias_gat, denom, maxenc, N, NC);
    zero_stats_kernel<<<(8 * C128 + T - 1) / T, T, 0, stream>>>(stats, 8 * C128);

    // 1) xl = x @ W   (WMMA f32)
    gemm_nc128_kernel<<<N / 16, T, 0, stream>>>(x, W, nullptr, xl, N);

    // 2) per-node attention scores
    attn_scores_kernel<<<((size_t)N * 32 + T - 1) / T, T, 0, stream>>>(xl, att_src, att_dst, a_src, a_dst, N);

    // 3) segment softmax over incoming edges
    edge_max_kernel<<<(etot + T - 1) / T, T, 0, stream>>>(srcI, dstI, a_src, a_dst, e_val, maxenc, E, etot);
    edge_exp_kernel<<<(etot + T - 1) / T, T, 0, stream>>>(dstI, maxenc, e_val, denom, E, etot);

    // 4) scatter-aggregate messages into buf1 (pre-seeded with bias_gat)
    scatter_kernel<<<((long long)etot * 32 + T - 1) / T, T, 0, stream>>>(srcI, dstI, e_val, denom, xl, buf1, E, etot);

    // 5) MLP layer 1: buf2 = buf1 @ W1 + b1 ; BN -> ReLU -> buf1
    gemm_nc128_kernel<<<N / 16, T, 0, stream>>>(buf1, W1, b1, buf2, N);
    bn_stats_kernel<<<256, T, 0, stream>>>(buf2, sum1, sq1, N);
    bn_finalize_kernel<<<1, C128, 0, stream>>>(sum1, sq1, g1, be1, scale1, shift1, N);
    bn_apply_kernel<<<(NC + T - 1) / T, T, 0, stream>>>(buf2, scale1, shift1, nullptr, buf1, NC);

    // 6) MLP layer 2: buf2 = buf1 @ W2 + b2 ; BN -> ReLU -> + x -> out
    gemm_nc128_kernel<<<N / 16, T, 0, stream>>>(buf1, W2, b2, buf2, N);
    bn_stats_kernel<<<256, T, 0, stream>>>(buf2, sum2, sq2, N);
    bn_finalize_kernel<<<1, C128, 0, stream>>>(sum2, sq2, g2, be2, scale2, shift2, N);
    bn_apply_kernel<<<(NC + T - 1) / T, T, 0, stream>>>(buf2, scale2, shift2, x, out, NC);
}